// coef_to_array_64501818851635
// MI455X (gfx1250) — compile-verified
//
#include <hip/hip_runtime.h>
#include <stdint.h>

#define ITERS 100
#define BATCH 200000
#define TPB   32                 // one wave32 per workgroup -> scalar TDM descriptor
#define ROW_F 202                // floats per batch element in output (101 pts * 2)
#define TILE_F (TPB * ROW_F)     // 6464 floats staged per block
#define TILE_U8 ((TILE_F * 4) / 8) // 3232 eight-byte units (25856 B, contiguous)

typedef __attribute__((ext_vector_type(4))) unsigned int u32x4;
typedef __attribute__((ext_vector_type(8))) int          i32x8;
typedef __attribute__((ext_vector_type(4))) int          i32x4;
typedef __attribute__((ext_vector_type(4))) float        f32x4;

__global__ __launch_bounds__(TPB) void ifs_traj_kernel(
    const float* __restrict__ coef, const float* __restrict__ h,
    float* __restrict__ out) {
  __shared__ __align__(16) float buf[TILE_F];

  const int tid = threadIdx.x;
  const int b   = blockIdx.x * TPB + tid;

  // 12 coefficients per element: three coalesced 16B loads (48 B, 16-aligned).
  // Read-once stream -> non-temporal so it doesn't churn L2.
  const f32x4* cp = (const f32x4*)(coef + (size_t)b * 12);
  const f32x4 u = __builtin_nontemporal_load(cp + 0);
  const f32x4 v = __builtin_nontemporal_load(cp + 1);
  const f32x4 w = __builtin_nontemporal_load(cp + 2);
  const float a0 = u[0], a1 = u[1], a2 = u[2], a3 = u[3], a4 = v[0], a5 = v[1];
  const float q0 = v[2], q1 = v[3], q2 = w[0], q3 = w[1], q4 = w[2], q5 = w[3];

  const float j0 = fabsf(a0 * a3 - a1 * a2);
  const float j1 = fabsf(q0 * q3 - q1 * q2);
  const float p  = j0 / (j0 + j1);

  float x = 0.05f, y = 0.05f;
  // LDS row for this thread is a byte-exact image of out[b*202 ...]
  float2* row = (float2*)&buf[tid * ROW_F];
  row[0] = make_float2(0.05f, 0.05f);

  const float* hp = h + b;   // h[t*BATCH + b]: coalesced per-wave stream
#pragma unroll 4
  for (int t = 0; t < ITERS; ++t) {
    const float ht = __builtin_nontemporal_load(&hp[(size_t)t * BATCH]);
    if (t + 16 < ITERS)                     // global_prefetch_b8, 16 rows ahead
      __builtin_prefetch(&hp[(size_t)(t + 16) * BATCH], 0, 1);
    const bool  m  = ht > p;
    const float c0 = m ? q0 : a0;
    const float c1 = m ? q1 : a1;
    const float c2 = m ? q2 : a2;
    const float c3 = m ? q3 : a3;
    const float c4 = m ? q4 : a4;
    const float c5 = m ? q5 : a5;
    const float xn = c0 * x + c1 * y + c4;
    const float yn = c2 * x + c3 * y + c5;
    x = xn;
    y = yn;
    row[t + 1] = make_float2(xn, yn);       // conflict-free ds_store_b64
  }

  // Per-lane LDS stores must land before the TDM engine reads LDS.
  asm volatile("s_wait_dscnt 0" ::: "memory");

  // ---- Tensor Data Mover: 1-D bulk store, LDS tile -> contiguous global ----
  const uint64_t gaddr = (uint64_t)(uintptr_t)(out + (size_t)blockIdx.x * TILE_F);
  const uint32_t laddr = (uint32_t)(uintptr_t)buf;   // low 32 bits = LDS offset

  u32x4 g0;
  g0[0] = 1u;                                   // count=1, user descriptor
  g0[1] = laddr;                                // lds_addr (bytes)
  g0[2] = (uint32_t)gaddr;                      // global_addr[31:0]
  g0[3] = (uint32_t)((gaddr >> 32) & 0x1FFFFFFull) | (2u << 30); // addr[56:32] | type=2

  i32x8 g1;
  g1[0] = (int)(3u << 16);                      // data_size = 8 bytes
  g1[1] = (int)((TILE_U8 & 0xFFFFu) << 16);     // tensor_dim0[15:0] (bits 63:48)
  g1[2] = (int)(((uint32_t)TILE_U8 >> 16) | (1u << 16)); // tensor_dim0[31:16] | tensor_dim1=1
  g1[3] = (int)((uint32_t)TILE_U8 << 16);       // tile_dim0 = 3232 units (bits 127:112)
  g1[4] = 0;                                    // tile_dim1/tile_dim2 unused
  g1[5] = (int)TILE_U8;                         // tensor_dim0_stride[31:0]
  g1[6] = 0;
  g1[7] = 0;

  i32x4 gz = {0, 0, 0, 0};                      // groups 2/3: <=2-D tensor
#if defined(__clang_major__) && (__clang_major__ >= 23)
  i32x8 gz8 = {0, 0, 0, 0, 0, 0, 0, 0};
  __builtin_amdgcn_tensor_store_from_lds(g0, g1, gz, gz, gz8, 0);
#else
  __builtin_amdgcn_tensor_store_from_lds(g0, g1, gz, gz, 0);
#endif
  __builtin_amdgcn_s_wait_tensorcnt(0);
}

extern "C" void kernel_launch(void* const* d_in, const int* in_sizes, int n_in,
                              void* d_out, int out_size, void* d_ws, size_t ws_size,
                              hipStream_t stream) {
  const float* coef = (const float*)d_in[0];   // (BATCH, 12) f32
  const float* h    = (const float*)d_in[1];   // (ITERS, BATCH) f32
  float* out        = (float*)d_out;           // (BATCH, 101, 2) f32

  ifs_traj_kernel<<<dim3(BATCH / TPB), dim3(TPB), 0, stream>>>(coef, h, out);
}